// HardTripletLoss_3040836845690
// MI455X (gfx1250) — compile-verified
//
#include <hip/hip_runtime.h>
#include <math.h>

// ---------------------------------------------------------------------------
// HardTripletLoss mining, CDNA5 (gfx1250, wave32).
// d2[row] = ||x_row - anchor||^2 computed 16 rows at a time via
// V_WMMA_F32_16X16X4_F32 (A == B == diff chunk => C diagonal = partial d2).
// Streaming-bound: 205 MB @ 23.3 TB/s ~ 8.8 us floor; WMMA compute is free.
// ---------------------------------------------------------------------------

typedef __attribute__((ext_vector_type(2))) float v2f;
typedef __attribute__((ext_vector_type(8))) float v8f;

#define DIM 256
#define MARGIN_F 23.0f
#define MARGIN2_F 529.0f
#define SENT_MAX (-1.0f)
#define SENT_MIN (3.4e38f)

__global__ void htl_init_ws(int* ws) {
  ws[0] = __float_as_int(SENT_MAX);   // masked-max accumulator (positives)
  ws[1] = __float_as_int(SENT_MIN);   // min accumulator (negatives)
}

// mode 0: positives -> masked max of d2 (d2 < MARGIN^2), atomicMax into ws[0]
// mode 1: negatives -> min of d2, atomicMin into ws[1]
__global__ __launch_bounds__(256) void htl_tile_dist(
    const float* __restrict__ anchor,
    const float* __restrict__ X,
    int nrows, int* __restrict__ ws, int mode) {
  __shared__ float sA[DIM];
  __shared__ float sRed[8];

  const int tid = threadIdx.x;
  sA[tid] = anchor[tid];              // stage anchor in LDS once per block
  __syncthreads();

  const int lane = tid & 31;
  const int wave = tid >> 5;
  const int wavesPerBlock = blockDim.x >> 5;        // 8
  const int gWave  = blockIdx.x * wavesPerBlock + wave;
  const int nWaves = gridDim.x * wavesPerBlock;
  const int nTiles = (nrows + 15) >> 4;

  const int  mrow   = lane & 15;                    // row-in-tile this lane loads
  const int  kOff   = (lane >= 16) ? 2 : 0;         // K sub-offset per A/B layout
  const bool isDiag = (lane < 8) || (lane >= 24);   // lanes holding C[m][m]
  const int  diagIdx = (lane < 8) ? lane : (lane - 24);

  const float sentinel = (mode == 0) ? SENT_MAX : SENT_MIN;
  float best = sentinel;

  for (int t = gWave; t < nTiles; t += nWaves) {
    const int  row   = t * 16 + mrow;
    const bool valid = (row < nrows);
    const int  rowc  = valid ? row : (nrows - 1);   // clamp addr, mask value later
    const float* __restrict__ xrow = X + (size_t)rowc * DIM;

    // Prefetch next tile's row for this lane (wave-uniform branch).
    const int tn = t + nWaves;
    if (tn < nTiles) {
      int prow = tn * 16 + mrow;
      if (prow >= nrows) prow = nrows - 1;
      __builtin_prefetch(X + (size_t)prow * DIM, 0, 1);
    }

    v8f acc = {};
#pragma unroll 8
    for (int k0 = 0; k0 < DIM; k0 += 4) {
      const int k = k0 + kOff;
      v2f xv = *(const v2f*)(xrow + k);             // global b64, per-lane
      v2f av = *(const v2f*)(&sA[k]);               // ds b64 (addr uniform per half-wave)
      v2f dv = xv - av;
      // C += diff * diff^T : A and B share the same VGPR pair for 16x16x4 f32.
      acc = __builtin_amdgcn_wmma_f32_16x16x4_f32(
          false, dv, false, dv, (short)0, acc, false, false);
    }

    // Extract this lane's diagonal element C[m][m] (d2 for row m).
    float d2 = acc[0];
#pragma unroll
    for (int j = 1; j < 8; ++j) d2 = (diagIdx == j) ? acc[j] : d2;

    float v;
    if (mode == 0) {
      v = (isDiag && valid && (d2 < MARGIN2_F)) ? d2 : SENT_MAX;
      best = fmaxf(best, v);
    } else {
      v = (isDiag && valid) ? d2 : SENT_MIN;
      best = fminf(best, v);
    }
  }

  // Wave32 tree reduction.
#pragma unroll
  for (int off = 16; off > 0; off >>= 1) {
    float o = __shfl_xor(best, off, 32);
    best = (mode == 0) ? fmaxf(best, o) : fminf(best, o);
  }
  if (lane == 0) sRed[wave] = best;
  __syncthreads();

  if (tid == 0) {
    float r = sRed[0];
    for (int w = 1; w < wavesPerBlock; ++w)
      r = (mode == 0) ? fmaxf(r, sRed[w]) : fminf(r, sRed[w]);
    // f32 bit pattern order == int order here: values are >= 0 or the single
    // negative sentinel -1.0f, so int atomics implement float max/min.
    if (mode == 0) atomicMax(ws + 0, __float_as_int(r));
    else           atomicMin(ws + 1, __float_as_int(r));
  }
}

// One wave: fallback d2 of positives row 0 (reference argmax default when no
// positive is under the margin), then the scalar loss.
__global__ __launch_bounds__(32) void htl_finalize(
    const float* __restrict__ anchor,
    const float* __restrict__ pos,
    const int* __restrict__ ws,
    float* __restrict__ out) {
  const int lane = threadIdx.x;
  float s = 0.0f;
#pragma unroll
  for (int j = 0; j < 8; ++j) {
    float d = pos[lane * 8 + j] - anchor[lane * 8 + j];
    s += d * d;
  }
#pragma unroll
  for (int off = 16; off > 0; off >>= 1) s += __shfl_xor(s, off, 32);

  if (lane == 0) {
    float maxd2 = __int_as_float(ws[0]);
    if (maxd2 < 0.0f) maxd2 = s;                    // no positive under margin
    float mind2 = __int_as_float(ws[1]);
    float loss = sqrtf(maxd2) - sqrtf(mind2) + MARGIN_F;
    out[0] = fmaxf(loss, 0.0f);
  }
}

extern "C" void kernel_launch(void* const* d_in, const int* in_sizes, int n_in,
                              void* d_out, int out_size, void* d_ws, size_t ws_size,
                              hipStream_t stream) {
  const float* anchor = (const float*)d_in[0];
  const float* pos    = (const float*)d_in[1];
  const float* neg    = (const float*)d_in[2];
  float* out = (float*)d_out;
  int* ws = (int*)d_ws;

  const int nPos = in_sizes[1] / DIM;
  const int nNeg = in_sizes[2] / DIM;

  htl_init_ws<<<1, 1, 0, stream>>>(ws);

  const int blocks = 512;   // 4096 waves for ~6250 tiles/pool: saturates HBM
  htl_tile_dist<<<blocks, 256, 0, stream>>>(anchor, pos, nPos, ws, 0);
  htl_tile_dist<<<blocks, 256, 0, stream>>>(anchor, neg, nNeg, ws, 1);

  htl_finalize<<<1, 32, 0, stream>>>(anchor, pos, ws, out);
}